// GCN_63995012710872
// MI455X (gfx1250) — compile-verified
//
#include <hip/hip_runtime.h>

#define N_NODES 100000
#define N_EDGES 3200000
#define D_IN    128
#define D_HID   64
#define D_OUT   10

typedef __attribute__((ext_vector_type(2))) float v2f;
typedef __attribute__((ext_vector_type(8))) float v8f;

// ---------------- small utility kernels ----------------

__global__ void set_ones_f32(float* __restrict__ p, int n) {
  int i = blockIdx.x * blockDim.x + threadIdx.x;
  if (i < n) p[i] = 1.0f;
}

__global__ void zero_f32(float* __restrict__ p, long long n) {
  long long i = (long long)blockIdx.x * blockDim.x + threadIdx.x;
  if (i < n) p[i] = 0.0f;
}

__global__ void count_deg(const int* __restrict__ dst, float* __restrict__ deg, int e) {
  int i = blockIdx.x * blockDim.x + threadIdx.x;
  if (i < e) atomicAdd(&deg[dst[i]], 1.0f);
}

__global__ void rsqrt_inplace(float* __restrict__ p, int n) {
  int i = blockIdx.x * blockDim.x + threadIdx.x;
  if (i < n) p[i] = rsqrtf(p[i]);
}

// ---------------- B-operand pre-pack ----------------
// Rearrange W[Din x Dout] into the V_WMMA_F32_16X16X4_F32 B lane layout so the
// GEMM inner loop does one coalesced 8B load per tile with no bounds guard.
// Element (ks, lane, t) holds {W[ka][col], W[ka+1][col]} with
//   ka  = ks*4 + (lane>>4)*2     (lanes 16-31 carry K rows 2,3)
//   col = t*16 + (lane&15)       (zero-padded past Dout)
__global__ void pack_B(const float* __restrict__ W, float* __restrict__ Bpk,
                       int Din, int Dout, int NT) {
  int idx = blockIdx.x * blockDim.x + threadIdx.x;
  int total = (Din / 4) * 32 * NT;
  if (idx >= total) return;
  int t    = idx % NT;
  int lane = (idx / NT) % 32;
  int ks   = idx / (NT * 32);
  int ka   = ks * 4 + (lane >> 4) * 2;
  int col  = t * 16 + (lane & 15);
  float bx = 0.0f, by = 0.0f;
  if (col < Dout) {
    bx = W[ka * Dout + col];
    by = W[(ka + 1) * Dout + col];
  }
  Bpk[2 * idx]     = bx;
  Bpk[2 * idx + 1] = by;
}

// ---------------- WMMA GEMM: Hp[i,n] = dinv[i] * sum_k X[i,k]*W[k,n] ----------------
// One wave per 16-row tile; NT independent 16-wide N-tiles per wave (reuses the
// A fragment NT times, 4 independent WMMA chains). All shapes compile-time so
// the K loop is branch-free: 1 A load + NT B loads + NT wmma per K-step.

template <int NT, int DIN, int NVALID, int NSTORE>
__global__ __launch_bounds__(32)
void gcn_wmma_gemm(const float* __restrict__ X, const float* __restrict__ Bpk,
                   const float* __restrict__ dinv, float* __restrict__ Hp) {
  const int lane = threadIdx.x & 31;
  const int half = lane >> 4;
  const int l16  = lane & 15;

  const int mBase = blockIdx.x << 4;
  const int rowA  = mBase + l16;
  const float* xr = X + (size_t)rowA * DIN + half * 2;   // A: {K+0,K+1} / {K+2,K+3}
  const v2f*   bp = (const v2f*)Bpk + (size_t)lane * NT; // packed B base for this lane

  v8f acc[NT] = {};
  constexpr int KS = DIN / 4;

  for (int ks = 0; ks < KS; ++ks) {
    v2f a = *(const v2f*)(xr + ks * 4);
#pragma unroll
    for (int t = 0; t < NT; ++t) {
      v2f b = bp[(size_t)ks * 32 * NT + t];
      // 8 args: (neg_a, A, neg_b, B, c_mod, C, reuse_a, reuse_b)
      acc[t] = __builtin_amdgcn_wmma_f32_16x16x4_f32(
          false, a, false, b, (short)0, acc[t], false, false);
    }
  }

  // Epilogue: scale rows by dinv and store (store mask constant-folds for NVALID=64)
  const int r0 = mBase + half * 8;
#pragma unroll
  for (int t = 0; t < NT; ++t) {
    const int col = t * 16 + l16;
    if (col < NVALID) {
#pragma unroll
      for (int v = 0; v < 8; ++v) {
        const int r = r0 + v;
        Hp[(size_t)r * NSTORE + col] = dinv[r] * acc[t][v];
      }
    }
  }
}

// ---------------- edge scatter: Agg[dst,f] += Hp[src,f] ----------------

// 64-wide layers: 2 features/thread, one b64 load + 2 f32 atomics.
__global__ void scatter_add64(const float* __restrict__ Hp, const int* __restrict__ src,
                              const int* __restrict__ dst, float* __restrict__ agg, int E) {
  long long idx = (long long)blockIdx.x * blockDim.x + threadIdx.x;
  if (idx >= (long long)E * 32) return;
  int e = (int)(idx >> 5);
  int f = ((int)idx & 31) * 2;
  int s = src[e], d = dst[e];
  v2f v = *(const v2f*)(Hp + (size_t)s * 64 + f);
  float* p = agg + (size_t)d * 64 + f;
  atomicAdd(p,     v.x);
  atomicAdd(p + 1, v.y);
}

// Generic scalar scatter (used for D=10 output layer).
template <int D>
__global__ void scatter_add(const float* __restrict__ Hp, const int* __restrict__ src,
                            const int* __restrict__ dst, float* __restrict__ agg, int E) {
  long long idx = (long long)blockIdx.x * blockDim.x + threadIdx.x;
  if (idx >= (long long)E * D) return;
  int e = (int)(idx / D);
  int f = (int)(idx - (long long)e * D);
  float v = Hp[(size_t)src[e] * D + f];
  atomicAdd(agg + (size_t)dst[e] * D + f, v);
}

// ---------------- epilogue: out = act(dinv[i]*(Agg + Hp_selfloop) + bias) ----------------

template <int D, bool RELU>
__global__ void gcn_epilogue(const float* __restrict__ agg, const float* __restrict__ Hp,
                             const float* __restrict__ dinv, const float* __restrict__ bias,
                             float* __restrict__ out, int n) {
  long long idx = (long long)blockIdx.x * blockDim.x + threadIdx.x;
  if (idx >= (long long)n * D) return;
  int i = (int)(idx / D);
  int f = (int)(idx - (long long)i * D);
  float v = dinv[i] * (agg[idx] + Hp[idx]) + bias[f];
  if (RELU) v = fmaxf(v, 0.0f);
  out[idx] = v;
}

// ---------------- host orchestration ----------------

extern "C" void kernel_launch(void* const* d_in, const int* in_sizes, int n_in,
                              void* d_out, int out_size, void* d_ws, size_t ws_size,
                              hipStream_t stream) {
  const float* x  = (const float*)d_in[0];
  const int*   ei = (const int*)d_in[1];
  const float* W1 = (const float*)d_in[2];
  const float* b1 = (const float*)d_in[3];
  const float* W2 = (const float*)d_in[4];
  const float* b2 = (const float*)d_in[5];
  const float* W3 = (const float*)d_in[6];
  const float* b3 = (const float*)d_in[7];
  float* out = (float*)d_out;

  const int* src = ei;            // edge_index[0]
  const int* dst = ei + N_EDGES;  // edge_index[1]

  // workspace layout (~77 MB + 32 KB)
  float* dinv = (float*)d_ws;                       // N
  float* Hp   = dinv + N_NODES;                     // N*64
  float* Agg  = Hp + (size_t)N_NODES * D_HID;       // N*64
  float* Hout = Agg + (size_t)N_NODES * D_HID;      // N*64
  float* Bpk  = Hout + (size_t)N_NODES * D_HID;     // up to (128/4)*32*4*2 = 8192 floats

  const int T = 256;
  const int MT = N_NODES / 16;  // 6250, exact

  // --- symmetric normalization: dinv = rsqrt(1 + in-degree) ---
  set_ones_f32<<<(N_NODES + T - 1) / T, T, 0, stream>>>(dinv, N_NODES);
  count_deg<<<(N_EDGES + T - 1) / T, T, 0, stream>>>(dst, dinv, N_EDGES);
  rsqrt_inplace<<<(N_NODES + T - 1) / T, T, 0, stream>>>(dinv, N_NODES);

  // --- layer 1: x[100k,128] @ W1[128,64], relu ---
  {
    long long n  = (long long)N_NODES * D_HID;
    long long te = (long long)N_EDGES * 32;
    int pk = (D_IN / 4) * 32 * 4;
    zero_f32<<<(int)((n + T - 1) / T), T, 0, stream>>>(Agg, n);
    pack_B<<<(pk + T - 1) / T, T, 0, stream>>>(W1, Bpk, D_IN, D_HID, 4);
    gcn_wmma_gemm<4, D_IN, D_HID, D_HID><<<MT, 32, 0, stream>>>(x, Bpk, dinv, Hp);
    scatter_add64<<<(int)((te + T - 1) / T), T, 0, stream>>>(Hp, src, dst, Agg, N_EDGES);
    gcn_epilogue<D_HID, true><<<(int)((n + T - 1) / T), T, 0, stream>>>(
        Agg, Hp, dinv, b1, Hout, N_NODES);
  }

  // --- layer 2: h1[100k,64] @ W2[64,64], relu ---
  {
    long long n  = (long long)N_NODES * D_HID;
    long long te = (long long)N_EDGES * 32;
    int pk = (D_HID / 4) * 32 * 4;
    zero_f32<<<(int)((n + T - 1) / T), T, 0, stream>>>(Agg, n);
    pack_B<<<(pk + T - 1) / T, T, 0, stream>>>(W2, Bpk, D_HID, D_HID, 4);
    gcn_wmma_gemm<4, D_HID, D_HID, D_HID><<<MT, 32, 0, stream>>>(Hout, Bpk, dinv, Hp);
    scatter_add64<<<(int)((te + T - 1) / T), T, 0, stream>>>(Hp, src, dst, Agg, N_EDGES);
    gcn_epilogue<D_HID, true><<<(int)((n + T - 1) / T), T, 0, stream>>>(
        Agg, Hp, dinv, b2, Hout, N_NODES);
  }

  // --- layer 3: h2[100k,64] @ W3[64,10], no relu -> d_out ---
  {
    long long n  = (long long)N_NODES * D_OUT;
    long long te = (long long)N_EDGES * D_OUT;
    int pk = (D_HID / 4) * 32 * 1;
    zero_f32<<<(int)((n + T - 1) / T), T, 0, stream>>>(Agg, n);
    pack_B<<<(pk + T - 1) / T, T, 0, stream>>>(W3, Bpk, D_HID, D_OUT, 1);
    gcn_wmma_gemm<1, D_HID, D_OUT, D_OUT><<<MT, 32, 0, stream>>>(Hout, Bpk, dinv, Hp);
    scatter_add<D_OUT><<<(int)((te + T - 1) / T), T, 0, stream>>>(Hp, src, dst, Agg, N_EDGES);
    gcn_epilogue<D_OUT, false><<<(int)((n + T - 1) / T), T, 0, stream>>>(
        Agg, Hp, dinv, b3, out, N_NODES);
  }
}